// YOWOMultiTask_34196529611123
// MI455X (gfx1250) — compile-verified
//
#include <hip/hip_runtime.h>
#include <hip/hip_bf16.h>
#include <stdint.h>

// ---------------------------------------------------------------------------
// YOWO multi-task post-process for MI455X (gfx1250, wave32).
//
// Dataflow: top-50-by-conf selection per (batch, level) row, then activations
// only on the 150 selected anchors per batch. Traffic ~13MB vs ~1GB naive.
//
// CDNA5 paths:
//   * Tensor Data Mover: tensor_load_to_lds DMAs the contiguous conf row
//     (up to 100KB) into LDS; completion via s_wait_tensorcnt (TENSORcnt).
//   * wave32 shuffle argmax reductions (2 barriers/iter instead of 11).
//   * global_prefetch_b8 to overlap gather-row fetches with softmax math.
// No WMMA: workload is selection/gather/elementwise — zero matmul content.
// ---------------------------------------------------------------------------

#define BATCH   32
#define TOPK    50
#define N_OBJ   36
#define N_ACT   157
#define N_REL   26
#define N_CH    225      // 4 box + 1 conf + 1 inter + 36 + 157 + 26
#define M0      25600    // (1280/8)^2
#define M1      6400     // (1280/16)^2
#define M2      1600     // (1280/32)^2
#define MMAX    M0
#define NROWS   (BATCH * 3 * TOPK)   // 4800

// ---- TDM availability / arity probing -------------------------------------
#if defined(__has_builtin)
#  if __has_builtin(__builtin_amdgcn_tensor_load_to_lds)
#    if defined(__has_include)
#      if __has_include(<hip/amd_detail/amd_gfx1250_TDM.h>)
#        define YOWO_TDM 2   /* therock / clang-23: 6-arg builtin */
#      else
#        define YOWO_TDM 1   /* ROCm 7.2 / clang-22: 5-arg builtin */
#      endif
#    else
#      define YOWO_TDM 1
#    endif
#  endif
#endif
#ifndef YOWO_TDM
#  define YOWO_TDM 0
#endif

typedef unsigned int v4u __attribute__((ext_vector_type(4)));
typedef int          v4i __attribute__((ext_vector_type(4)));
typedef int          v8i __attribute__((ext_vector_type(8)));

__device__ __forceinline__ float sigmoidf_fast(float x) {
    return 1.0f / (1.0f + __expf(-x));
}

// wave32 argmax with lowest-index tie-break (matches lax.top_k ordering)
__device__ __forceinline__ void wave_argmax(float& v, int& i) {
    #pragma unroll
    for (int off = 16; off; off >>= 1) {
        float ov = __shfl_xor(v, off, 32);
        int   oi = __shfl_xor(i, off, 32);
        if (ov > v || (ov == v && oi < i)) { v = ov; i = oi; }
    }
}

// ---------------------------------------------------------------------------
// Kernel 1: per-(batch,level) top-50 over raw conf (sigmoid is monotone).
// grid = (32, 3); block = 1024 (32 waves).  Dynamic LDS:
//   [0, MMAX)         staged conf values (float)
//   [MMAX, MMAX+32)   per-wave winner values
//   [MMAX+32, +64)    per-wave winner indices
// ---------------------------------------------------------------------------
__global__ __launch_bounds__(1024)
void yowo_topk(const float* __restrict__ conf0,
               const float* __restrict__ conf1,
               const float* __restrict__ conf2,
               int* __restrict__ tk_idx, float* __restrict__ tk_val)
{
    const int b  = blockIdx.x;
    const int lv = blockIdx.y;
    const int M  = (lv == 0) ? M0 : (lv == 1) ? M1 : M2;
    const float* confp = (lv == 0) ? conf0 : (lv == 1) ? conf1 : conf2;
    const float* __restrict__ src = confp + (size_t)b * M;

    extern __shared__ float smem[];
    float* vals = smem;                      // [M] conf values
    float* redv = smem + MMAX;               // [32]
    int*   redi = (int*)(smem + MMAX + 32);  // [32]
    const int tid  = threadIdx.x;
    const int lane = tid & 31;
    const int wid  = tid >> 5;

#if YOWO_TDM
    // --- DMA the conf row into LDS with the Tensor Data Mover -------------
    if (tid == 0) {
        unsigned long long ga = (unsigned long long)(uintptr_t)src;
        unsigned int lds_off  = (unsigned int)(uintptr_t)vals; // LDS flat low 32b = byte offset

        // D# group 0: count=1 | lds_addr | global_addr[56:0] | type=2
        v4u g0;
        g0.x = 1u;
        g0.y = lds_off;
        g0.z = (unsigned int)ga;
        g0.w = (unsigned int)((ga >> 32) & 0x01FFFFFFu) | (2u << 30);

        // D# group 1: data_size=4B; tensor_dim0=M; tensor_dim1=1;
        //             tile_dim0=M (<=65535); tile_dim1=1; stride0=M.
        v8i g1;
        g1.s0 = (2 << 16);                           // data_size = 2 -> 4 bytes
        g1.s1 = (int)((unsigned)(M & 0xFFFF) << 16); // tensor_dim0[15:0] @ [63:48]
        g1.s2 = ((M >> 16) & 0xFFFF) | (1 << 16);    // tensor_dim0[31:16] | tensor_dim1=1
        g1.s3 = (int)((unsigned)(M & 0xFFFF) << 16); // tile_dim0 = M @ [127:112]
        g1.s4 = 1;                                   // tile_dim1 = 1, tile_dim2 = 0
        g1.s5 = M;                                   // tensor_dim0_stride[31:0]
        g1.s6 = 0;
        g1.s7 = 0;

        v4i zero4 = {0, 0, 0, 0};
#if YOWO_TDM == 2
        v8i zero8 = {0, 0, 0, 0, 0, 0, 0, 0};
        __builtin_amdgcn_tensor_load_to_lds(g0, g1, zero4, zero4, zero8, 0);
#else
        __builtin_amdgcn_tensor_load_to_lds(g0, g1, zero4, zero4, 0);
#endif
        __builtin_amdgcn_s_wait_tensorcnt(0);
    }
    __syncthreads();
#else
    // --- fallback: cooperative float4 staging -----------------------------
    {
        const float4* __restrict__ src4 = (const float4*)src;
        float4* vals4 = (float4*)vals;
        const int M4 = M >> 2;
        for (int m = tid; m < M4; m += 1024) vals4[m] = src4[m];
    }
    __syncthreads();
#endif

    // --- 50 argmax passes; float4 LDS scans + wave shuffle reduction -------
    const int out_base = (b * 3 + lv) * TOPK;
    const float4* vals4 = (const float4*)vals;
    const int M4 = M >> 2;

    for (int k = 0; k < TOPK; ++k) {
        float best = -__builtin_inff();
        int   bi   = 0x7fffffff;
        for (int m4 = tid; m4 < M4; m4 += 1024) {
            float4 v = vals4[m4];                 // ds_load_b128
            float lb = v.x; int lj = 0;           // strict '>' keeps lowest index
            if (v.y > lb) { lb = v.y; lj = 1; }
            if (v.z > lb) { lb = v.z; lj = 2; }
            if (v.w > lb) { lb = v.w; lj = 3; }
            if (lb > best) { best = lb; bi = (m4 << 2) | lj; }
        }
        wave_argmax(best, bi);                    // intra-wave, no barriers
        if (lane == 0) { redv[wid] = best; redi[wid] = bi; }
        __syncthreads();
        if (tid < 32) {                           // wave 0 reduces 32 winners
            float wv = redv[lane];
            int   wi = redi[lane];
            wave_argmax(wv, wi);
            if (lane == 0) {
                tk_idx[out_base + k] = wi;
                tk_val[out_base + k] = sigmoidf_fast(wv);
                vals[wi] = -__builtin_inff();     // knock out winner
            }
        }
        __syncthreads();
    }
}

// ---------------------------------------------------------------------------
// Kernel 2: gather + activations. One wave32 per output row (4800 rows).
// block = 256 (8 waves), grid = 600.
// Channel map: [0:4) box, 4 conf, 5 inter, [6:42) obj, [42:199) act, [199:225) rel
// ---------------------------------------------------------------------------
__global__ __launch_bounds__(256)
void yowo_gather(const float* __restrict__ obj0, const float* __restrict__ act0,
                 const float* __restrict__ rel0, const float* __restrict__ int0,
                 const float* __restrict__ reg0,
                 const float* __restrict__ obj1, const float* __restrict__ act1,
                 const float* __restrict__ rel1, const float* __restrict__ int1,
                 const float* __restrict__ reg1,
                 const float* __restrict__ obj2, const float* __restrict__ act2,
                 const float* __restrict__ rel2, const float* __restrict__ int2,
                 const float* __restrict__ reg2,
                 const int* __restrict__ tk_idx, const float* __restrict__ tk_val,
                 float* __restrict__ out)
{
    const int row  = blockIdx.x * 8 + (threadIdx.x >> 5);
    const int lane = threadIdx.x & 31;
    if (row >= NROWS) return;

    const int b  = row / 150;
    const int r  = row - b * 150;
    const int lv = r / TOPK;
    const int k  = r - lv * TOPK;

    int M, n; float stride;
    const float *obj, *act, *rel, *itp, *reg;
    if (lv == 0) { M = M0; n = 160; stride =  8.f; obj = obj0; act = act0; rel = rel0; itp = int0; reg = reg0; }
    else if (lv == 1) { M = M1; n = 80; stride = 16.f; obj = obj1; act = act1; rel = rel1; itp = int1; reg = reg1; }
    else { M = M2; n = 40; stride = 32.f; obj = obj2; act = act2; rel = rel2; itp = int2; reg = reg2; }

    const int   t    = (b * 3 + lv) * TOPK + k;
    const int   idx  = tk_idx[t];
    const float cs   = tk_val[t];                 // sigmoid(conf), sorted desc
    const float keep = (cs > 0.05f) ? 1.0f : 0.0f;

    const size_t base = (size_t)b * M + (size_t)idx;
    float* __restrict__ orow = out + (size_t)row * N_CH;

    const float* op = obj + base * N_OBJ;
    const float* ap = act + base * N_ACT;
    const float* rp = rel + base * N_REL;
    const float* rg = reg + base * 4;

    // Prefetch the act/rel/reg cachelines so they overlap the softmax
    // reductions below (gfx1250 global_prefetch_b8).
    if (lane < 5) __builtin_prefetch(ap + (lane << 5), 0, 3);
    if (lane == 5) __builtin_prefetch(rp, 0, 3);
    if (lane == 6) __builtin_prefetch(rg, 0, 3);
    if (lane == 7) __builtin_prefetch(itp + base, 0, 3);

    // --- obj softmax over 36 channels (wave32 shuffle reductions) ---------
    {
        float o1 = op[lane];                                  // lane < 32 < 36
        float o2 = (lane < N_OBJ - 32) ? op[lane + 32] : -__builtin_inff();
        float m  = fmaxf(o1, o2);
        #pragma unroll
        for (int off = 16; off; off >>= 1) m = fmaxf(m, __shfl_xor(m, off, 32));
        float e1 = __expf(o1 - m);
        float e2 = (lane < N_OBJ - 32) ? __expf(o2 - m) : 0.0f;
        float s  = e1 + e2;
        #pragma unroll
        for (int off = 16; off; off >>= 1) s += __shfl_xor(s, off, 32);
        const float inv = keep / s;
        orow[6 + lane] = e1 * inv;
        if (lane < N_OBJ - 32) orow[6 + lane + 32] = e2 * inv;
    }

    // --- act sigmoid (157) -------------------------------------------------
    #pragma unroll
    for (int i = 0; i < 5; ++i) {
        int c = lane + (i << 5);
        if (c < N_ACT) orow[42 + c] = keep * sigmoidf_fast(ap[c]);
    }

    // --- rel sigmoid (26) --------------------------------------------------
    if (lane < N_REL) orow[199 + lane] = keep * sigmoidf_fast(rp[lane]);

    // --- box decode + conf + inter (lane 0) --------------------------------
    if (lane == 0) {
        const float ax = ((float)(idx % n) + 0.5f) * stride;
        const float ay = ((float)(idx / n) + 0.5f) * stride;
        const float cx = ax + rg[0] * stride;
        const float cy = ay + rg[1] * stride;
        const float hw = 0.5f * __expf(rg[2]) * stride;
        const float hh = 0.5f * __expf(rg[3]) * stride;
        orow[0] = (cx - hw) * keep;
        orow[1] = (cy - hh) * keep;
        orow[2] = (cx + hw) * keep;
        orow[3] = (cy + hh) * keep;
        orow[4] = cs * keep;
        orow[5] = keep * sigmoidf_fast(itp[base]);
    }
}

// ---------------------------------------------------------------------------
extern "C" void kernel_launch(void* const* d_in, const int* in_sizes, int n_in,
                              void* d_out, int out_size, void* d_ws, size_t ws_size,
                              hipStream_t stream)
{
    // setup_inputs order: level i -> conf, obj, act, rel, int, reg
    const float* conf0 = (const float*)d_in[0];
    const float* obj0  = (const float*)d_in[1];
    const float* act0  = (const float*)d_in[2];
    const float* rel0  = (const float*)d_in[3];
    const float* int0  = (const float*)d_in[4];
    const float* reg0  = (const float*)d_in[5];
    const float* conf1 = (const float*)d_in[6];
    const float* obj1  = (const float*)d_in[7];
    const float* act1  = (const float*)d_in[8];
    const float* rel1  = (const float*)d_in[9];
    const float* int1  = (const float*)d_in[10];
    const float* reg1  = (const float*)d_in[11];
    const float* conf2 = (const float*)d_in[12];
    const float* obj2  = (const float*)d_in[13];
    const float* act2  = (const float*)d_in[14];
    const float* rel2  = (const float*)d_in[15];
    const float* int2  = (const float*)d_in[16];
    const float* reg2  = (const float*)d_in[17];

    int*   tk_idx = (int*)d_ws;
    float* tk_val = (float*)((char*)d_ws + NROWS * sizeof(int));
    float* out    = (float*)d_out;

    const size_t shmem = (size_t)MMAX * sizeof(float)   // staged conf row
                       + 32 * sizeof(float)             // per-wave winner values
                       + 32 * sizeof(int);              // per-wave winner indices
    // Best-effort opt-in to >64KB dynamic LDS (gfx1250 WGP has 320KB).
    (void)hipFuncSetAttribute((const void*)yowo_topk,
                              hipFuncAttributeMaxDynamicSharedMemorySize,
                              (int)shmem);

    dim3 g1(BATCH, 3);
    yowo_topk<<<g1, 1024, shmem, stream>>>(conf0, conf1, conf2, tk_idx, tk_val);

    yowo_gather<<<NROWS / 8, 256, 0, stream>>>(
        obj0, act0, rel0, int0, reg0,
        obj1, act1, rel1, int1, reg1,
        obj2, act2, rel2, int2, reg2,
        tk_idx, tk_val, out);
}